// Critic_50826642981090
// MI455X (gfx1250) — compile-verified
//
#include <hip/hip_runtime.h>
#include <hip/hip_bf16.h>

#define TT   4096
#define BB   256
#define IND  128
#define HH   8
#define GDIM 32      // 4*H
#define NRST 32

typedef float v2f __attribute__((ext_vector_type(2)));
typedef float v8f __attribute__((ext_vector_type(8)));

// ---------------------------------------------------------------------------
// Kernel 0: reset mask[t] = (t in reset_idx)
// ---------------------------------------------------------------------------
__global__ void __launch_bounds__(256) reset_mask_kernel(
    const int* __restrict__ reset_idx, int* __restrict__ mask) {
  int t = blockIdx.x * blockDim.x + threadIdx.x;
  if (t >= TT) return;
  int m = 0;
#pragma unroll
  for (int i = 0; i < NRST; ++i) m |= (reset_idx[i] == t) ? 1 : 0;
  mask[t] = m;
}

// ---------------------------------------------------------------------------
// Kernel 1: xg[m, g] = sum_k x[m, k] * W_ih[g, k] + b_ih[g] + b_hh[g]
//   M = T*B rows, N = 32 gates, K = 128.  fp32 WMMA 16x16x4.
//   One wave -> one 16-row x 32-col tile (two 16x16 N-tiles sharing A).
// A frag (16x4 f32, 2 VGPR): lane<16 holds K={0,1}, lane>=16 K={2,3}, M=lane%16
// B frag (4x16 f32, 2 VGPR): lane<16 holds K={0,1}, lane>=16 K={2,3}, N=lane%16
// C/D (16x16 f32, 8 VGPR):   VGPR v -> M = v + 8*(lane/16), N = lane%16
// ---------------------------------------------------------------------------
__global__ void __launch_bounds__(256) xgates_wmma_kernel(
    const float* __restrict__ x,     // [T*B, 128]
    const float* __restrict__ W_ih,  // [32, 128]
    const float* __restrict__ b_ih,  // [32]
    const float* __restrict__ b_hh,  // [32]
    float* __restrict__ xg)          // [T*B, 32]
{
  const int lane  = threadIdx.x & 31;
  const int wave  = threadIdx.x >> 5;
  const long tile = (long)blockIdx.x * 8 + wave;  // 16-row tile
  const long m0   = tile * 16;
  const int  lh   = lane >> 4;   // 0/1 -> K pair select
  const int  lm   = lane & 15;   // M (for A) or N (for B/C)

  // accumulators pre-seeded with bias (bias depends only on N = lane%16)
  const float bias0 = b_ih[lm]      + b_hh[lm];
  const float bias1 = b_ih[16 + lm] + b_hh[16 + lm];
  v8f c0, c1;
#pragma unroll
  for (int v = 0; v < 8; ++v) { c0[v] = bias0; c1[v] = bias1; }

  const float* arow  = x    + (m0 + lm) * IND + 2 * lh;
  const float* brow0 = W_ih + lm        * IND + 2 * lh;
  const float* brow1 = W_ih + (16 + lm) * IND + 2 * lh;

#pragma unroll 4
  for (int k0 = 0; k0 < IND; k0 += 4) {
    v2f a  = *(const v2f*)(arow  + k0);   // 8B aligned: even float offsets
    v2f b0 = *(const v2f*)(brow0 + k0);
    v2f b1 = *(const v2f*)(brow1 + k0);
    c0 = __builtin_amdgcn_wmma_f32_16x16x4_f32(false, a, false, b0,
                                               (short)0, c0, false, false);
    c1 = __builtin_amdgcn_wmma_f32_16x16x4_f32(false, a, false, b1,
                                               (short)0, c1, false, false);
  }

  float* out = xg + m0 * GDIM;
#pragma unroll
  for (int v = 0; v < 8; ++v) {
    const int row = v + 8 * lh;
    out[row * GDIM + lm]      = c0[v];
    out[row * GDIM + 16 + lm] = c1[v];
  }
}

// ---------------------------------------------------------------------------
// Kernel 2: sequential LSTM scan + projection.
//   8 lanes per batch element; lane j owns h-index j (and its c), keeps a
//   replicated copy of all 8 h values for the 8-wide recurrence dots.
//   No inter-wave communication -> 64 single-wave workgroups spread over WGPs.
// ---------------------------------------------------------------------------
__device__ __forceinline__ float fast_sigmoid(float x) {
  return __builtin_amdgcn_rcpf(1.0f + __expf(-x));
}
__device__ __forceinline__ float fast_tanh(float x) {
  return fmaf(2.0f, fast_sigmoid(2.0f * x), -1.0f);
}

__global__ void __launch_bounds__(32) lstm_scan_kernel(
    const float* __restrict__ xg,      // [T*B, 32]
    const int*   __restrict__ mask,    // [T]
    const float* __restrict__ W_hh,    // [32, 8]  rows: i(0-7) f(8-15) g(16-23) o(24-31)
    const float* __restrict__ W_proj,  // [8]
    const float* __restrict__ b_proj,  // [1]
    float* __restrict__ value,         // [T*B]
    float* __restrict__ rnn_out)       // [T*B, 8]
{
  const int lane = threadIdx.x & 31;
  const int s    = lane >> 3;                  // batch slot within wave (0..3)
  const int j    = lane & 7;                   // h index owned by this lane
  const int b    = blockIdx.x * 4 + s;         // global batch element

  // recurrence weights for this lane's 4 gates (i,f,g,o at h-index j)
  float wi[8], wf[8], wg[8], wo[8], wpv[8];
#pragma unroll
  for (int k = 0; k < 8; ++k) {
    wi[k]  = W_hh[(0  + j) * HH + k];
    wf[k]  = W_hh[(8  + j) * HH + k];
    wg[k]  = W_hh[(16 + j) * HH + k];
    wo[k]  = W_hh[(24 + j) * HH + k];
    wpv[k] = W_proj[k];
  }
  const float bp = b_proj[0];

  float h[8];
#pragma unroll
  for (int k = 0; k < 8; ++k) h[k] = 0.0f;
  float c = 0.0f;

  const long base = (long)b * GDIM;

  // software-pipelined gate loads (t+1 in registers, t+8 prefetched)
  const float* g0 = xg + base;                 // t = 0
  float gi = g0[j], gf = g0[8 + j], gg = g0[16 + j], go = g0[24 + j];

  for (int t = 0; t < TT; ++t) {
    float ni = 0.f, nf = 0.f, ng = 0.f, no = 0.f;
    if (t + 1 < TT) {
      const float* gn = xg + (long)(t + 1) * BB * GDIM + base;
      ni = gn[j]; nf = gn[8 + j]; ng = gn[16 + j]; no = gn[24 + j];
    }
    if (t + 8 < TT)
      __builtin_prefetch(xg + (long)(t + 8) * BB * GDIM + base, 0, 0);

    if (mask[t]) {                             // uniform branch
      c = 0.0f;
#pragma unroll
      for (int k = 0; k < 8; ++k) h[k] = 0.0f;
    }

#pragma unroll
    for (int k = 0; k < 8; ++k) {
      gi = fmaf(wi[k], h[k], gi);
      gf = fmaf(wf[k], h[k], gf);
      gg = fmaf(wg[k], h[k], gg);
      go = fmaf(wo[k], h[k], go);
    }

    const float is = fast_sigmoid(gi);
    const float fs = fast_sigmoid(gf);
    const float os = fast_sigmoid(go);
    c = fmaf(fs, c, is * fast_tanh(gg));
    const float hj = os * fast_tanh(c);

    // re-share h across the 8-lane group (ds_bpermute, no barrier)
#pragma unroll
    for (int k = 0; k < 8; ++k) h[k] = __shfl(hj, k, 8);

    rnn_out[((long)t * BB + b) * HH + j] = hj;  // coalesced within group
    if (j == 0) {
      float v = bp;
#pragma unroll
      for (int k = 0; k < 8; ++k) v = fmaf(wpv[k], h[k], v);
      value[(long)t * BB + b] = v;
    }

    gi = ni; gf = nf; gg = ng; go = no;
  }
}

// ---------------------------------------------------------------------------
// launch
// ---------------------------------------------------------------------------
extern "C" void kernel_launch(void* const* d_in, const int* in_sizes, int n_in,
                              void* d_out, int out_size, void* d_ws, size_t ws_size,
                              hipStream_t stream) {
  const float* x      = (const float*)d_in[0];   // [T,B,128]
  const int*   ridx   = (const int*)  d_in[1];   // [32]
  const float* W_ih   = (const float*)d_in[2];   // [32,128]
  const float* W_hh   = (const float*)d_in[3];   // [32,8]
  const float* b_ih   = (const float*)d_in[4];   // [32]
  const float* b_hh   = (const float*)d_in[5];   // [32]
  const float* W_proj = (const float*)d_in[6];   // [8]
  const float* b_proj = (const float*)d_in[7];   // [1]

  float* value = (float*)d_out;                  // [T*B]
  float* rnn   = value + (size_t)TT * BB;        // [T*B,8]

  int*   mask = (int*)d_ws;                                   // 16 KB
  float* xg   = (float*)((char*)d_ws + 16384);                // 128 MB gates

  reset_mask_kernel<<<TT / 256, 256, 0, stream>>>(ridx, mask);

  // (T*B)/16 row-tiles, 8 waves (tiles) per 256-thread block
  const int tiles  = (TT * BB) / 16;             // 65536
  xgates_wmma_kernel<<<tiles / 8, 256, 0, stream>>>(x, W_ih, b_ih, b_hh, xg);

  lstm_scan_kernel<<<BB / 4, 32, 0, stream>>>(xg, mask, W_hh, W_proj, b_proj,
                                              value, rnn);
}